// AUV_39565238730960
// MI455X (gfx1250) — compile-verified
//
#include <hip/hip_runtime.h>
#include <math.h>

#define NCH    8
#define NBASIS 30
#define KPAD   32               // K padded to 32 for unguarded WMMA k-loop
#define NXS    256
#define NF     200
#define NXI    (NXS * NXS)      // 65536 complex elements per image
#define NXTOT  (NXI * 2)        // 131072 interleaved floats per image (the "n" axis)
#define NIMG   (NCH * NBASIS)   // 240 images

typedef float v2f __attribute__((ext_vector_type(2)));
typedef float v8f __attribute__((ext_vector_type(8)));

// ---------------------------------------------------------------------------
// Radix-2 Stockham autosort FFT core, N=256, 128 threads per row (t = 0..127).
// buf = this row's two ping-pong buffers. Natural order in/out.
// ---------------------------------------------------------------------------
__device__ __forceinline__ int fft256_stockham(float2 (*buf)[NXS], int t) {
    int cur = 0;
#pragma unroll
    for (int s = 0; s < 8; ++s) {
        const int m  = 1 << s;
        const int jm = (t >> s) << s;      // j*m
        const int k  = t & (m - 1);
        float2 a = buf[cur][t];
        float2 b = buf[cur][t + 128];
        float2 sum, dif;
        sum.x = a.x + b.x; sum.y = a.y + b.y;
        dif.x = a.x - b.x; dif.y = a.y - b.y;
        // w = exp(-2*pi*i * jm / 256)
        float ang = (float)jm * (-6.28318530717958647692f / 256.0f);
        float sn, cs;
        __sincosf(ang, &sn, &cs);
        float2 wd;
        wd.x = cs * dif.x - sn * dif.y;
        wd.y = cs * dif.y + sn * dif.x;
        const int i0 = k + (jm << 1);
        buf[1 - cur][i0]     = sum;
        buf[1 - cur][i0 + m] = wd;
        cur ^= 1;
        __syncthreads();
    }
    return cur;
}

// ---------------------------------------------------------------------------
// Pass 1: coil weighting + (-1)^(i+j) + row FFTs; write transposed scratch.
// grid = (128, 240), block = 256 (two rows per block, 128 threads per row).
// ---------------------------------------------------------------------------
__global__ __launch_bounds__(256) void fft_rows_kernel(
    const float* __restrict__ x, const float* __restrict__ csm,
    float* __restrict__ scratchT) {
    __shared__ float2 lds[2][2][NXS];   // [row-in-block][buffer][element]

    const int r   = threadIdx.x >> 7;   // 0 or 1: which row of the pair
    const int t   = threadIdx.x & 127;  // butterfly index
    const int img = blockIdx.y;         // 0..239
    const int c   = img / NBASIS;
    const int b   = img % NBASIS;
    const int i   = blockIdx.x * 2 + r; // row index in image

    const float* xrow = x   + ((size_t)b * NXI + (size_t)i * NXS) * 2;
    const float* crow = csm + ((size_t)c * NXI + (size_t)i * NXS) * 2;

#pragma unroll
    for (int e = 0; e < 2; ++e) {
        const int j  = t + e * 128;
        const float xr = xrow[2 * j], xi = xrow[2 * j + 1];
        const float cr = crow[2 * j], ci = crow[2 * j + 1];
        const float sg = ((i + j) & 1) ? -1.0f : 1.0f;   // ifftshift phase
        float2 v;
        v.x = sg * (xr * cr - xi * ci);
        v.y = sg * (xr * ci + xi * cr);
        lds[r][0][j] = v;
    }
    __syncthreads();

    const int cur = fft256_stockham(lds[r], t);

    float2* sc = (float2*)scratchT + (size_t)img * NXI;
#pragma unroll
    for (int e = 0; e < 2; ++e) {
        const int q = t + e * 128;
        sc[(size_t)q * NXS + i] = lds[r][cur][q];   // transposed write
    }
}

// ---------------------------------------------------------------------------
// Pass 2: column FFTs on transposed scratch; apply (-1)^(ki+kj)/256; write Yr
// into K-padded layout (c, 32, NXI) float2, i.e. row c*32+b.
// LDS fill uses the gfx1250 async global->LDS pipe (ASYNCcnt).
// ---------------------------------------------------------------------------
__global__ __launch_bounds__(256) void fft_cols_kernel(
    const float* __restrict__ scratchT, float* __restrict__ Yr) {
    __shared__ float2 lds[2][2][NXS];   // [row-in-block][buffer][element]

    const int r   = threadIdx.x >> 7;
    const int t   = threadIdx.x & 127;
    const int img = blockIdx.y;
    const int c   = img / NBASIS;
    const int b   = img % NBASIS;
    const int kj  = blockIdx.x * 2 + r;  // output column (pass-1 output row)

    const float2* sc = (const float2*)scratchT + (size_t)img * NXI + (size_t)kj * NXS;

    // One async b128 per thread: copies global elements 2t and 2t+1 into elements 2t and 2t+1 of this row's LDS buffer 0.
    {
        unsigned int lofs = (unsigned int)(uintptr_t)&lds[r][0][2 * t];
        unsigned long long ga = (unsigned long long)(const void*)(sc + 2 * t);
        asm volatile("global_load_async_to_lds_b128 %0, %1, off"
                     :: "v"(lofs), "v"(ga) : "memory");
        asm volatile("s_wait_asynccnt 0x0" ::: "memory");
    }
    __syncthreads();

    const int cur = fft256_stockham(lds[r], t);

    float2* yr = (float2*)Yr + (size_t)(c * KPAD + b) * NXI;
#pragma unroll
    for (int e = 0; e < 2; ++e) {
        const int ki = t + e * 128;
        float2 v = lds[r][cur][ki];
        const float sg = ((ki + kj) & 1) ? (-1.0f / 256.0f) : (1.0f / 256.0f);
        v.x *= sg; v.y *= sg;
        yr[(size_t)ki * NXS + kj] = v;
    }
}

// ---------------------------------------------------------------------------
// Zero the pad rows k=30,31 of each channel in Yr (re-done every launch so the
// result is deterministic regardless of workspace contents).
// grid = 2048, block = 256; one float4 per thread.
// ---------------------------------------------------------------------------
__global__ __launch_bounds__(256) void zero_pad_kernel(float* __restrict__ Yr) {
    const size_t idx     = (size_t)blockIdx.x * blockDim.x + threadIdx.x;
    const size_t per_row = NXTOT / 4;                    // float4 per padded row
    const int    row     = (int)(idx / per_row);         // 0..15
    const size_t off     = (idx % per_row) * 4;
    const int    c       = row >> 1;
    const int    k       = NBASIS + (row & 1);           // 30 or 31
    float4 z = make_float4(0.f, 0.f, 0.f, 0.f);
    *(float4*)(Yr + ((size_t)(c * KPAD + k)) * NXTOT + off) = z;
}

// ---------------------------------------------------------------------------
// Pass 3: out[c,f,n] = mask[f,n] * sum_k VT[k,f] * Yr[c,k,n]
// Per-channel GEMM, M=NF(200), N=NXTOT(131072), K=32 (zero-padded).
// WMMA f32 16x16x4, 8 unguarded k-steps. One wave per 16f x 16n tile,
// 8 waves per block covering 128 n. grid = (1024, 13, 8).
// ---------------------------------------------------------------------------
__global__ __launch_bounds__(256) void proj_mask_kernel(
    const float* __restrict__ Yr, const float* __restrict__ VT,
    const unsigned char* __restrict__ mask, float* __restrict__ out) {
    const int lane = threadIdx.x & 31;
    const int wave = threadIdx.x >> 5;
    const int half = lane >> 4;          // 0: K pair {0,1}; 1: K pair {2,3}
    const int ln   = lane & 15;

    const int c  = blockIdx.z;
    const int f0 = blockIdx.y * 16;
    const int n0 = blockIdx.x * 128 + wave * 16;
    const int n  = n0 + ln;

    // Prefetch the mask rows this tile will need (global_prefetch_b8 path).
    {
        int fpre = f0 + ln; if (fpre >= NF) fpre = NF - 1;
        __builtin_prefetch(mask + (size_t)fpre * NXTOT + n0, 0, 1);
    }

    // A operand: 16x4 tile of VT^T per k-step. lane = half*16 + m holds
    // {VT[k0][f0+m], VT[k0+1][f0+m]} with k0 = ks*4 + half*2.
    // Branchless zero-padding: clamp indices (loads always in-bounds), select 0.
    const int  f      = f0 + ln;
    const bool fvalid = (f < NF);
    const int  fc     = fvalid ? f : (NF - 1);
    v2f A[8];
#pragma unroll
    for (int ks = 0; ks < 8; ++ks) {
        const int k0  = ks * 4 + half * 2;
        const int kc0 = (k0     < NBASIS) ? k0     : 0;
        const int kc1 = (k0 + 1 < NBASIS) ? k0 + 1 : 0;
        float a0 = VT[(size_t)kc0 * NF + fc];
        float a1 = VT[(size_t)kc1 * NF + fc];
        a0 = (fvalid && k0     < NBASIS) ? a0 : 0.0f;
        a1 = (fvalid && k0 + 1 < NBASIS) ? a1 : 0.0f;
        A[ks] = (v2f){a0, a1};
    }

    v8f acc = {0.f, 0.f, 0.f, 0.f, 0.f, 0.f, 0.f, 0.f};
    const float* Bbase = Yr + (size_t)c * KPAD * NXTOT + n;
#pragma unroll
    for (int ks = 0; ks < 8; ++ks) {
        const int k0 = ks * 4 + half * 2;
        v2f Bv = (v2f){Bbase[(size_t)k0 * NXTOT], Bbase[(size_t)(k0 + 1) * NXTOT]};
        acc = __builtin_amdgcn_wmma_f32_16x16x4_f32(
            /*neg_a=*/false, A[ks], /*neg_b=*/false, Bv,
            /*c_mod=*/(short)0, acc, /*reuse_a=*/false, /*reuse_b=*/false);
    }

    // Epilogue: D VGPR rr holds rows m = rr (lanes 0-15) / 8+rr (lanes 16-31).
    // Base pointers computed once; per-row accesses use immediate offsets
    // (mask row stride 131072 B, out row stride 524288 B, both < 8 MB IOFFSET).
    // Non-temporal stores keep the L2-resident Yr/mask working set hot.
    const int mbase = f0 + half * 8;
    const unsigned char* mp = mask + (size_t)mbase * NXTOT + n;
    float* op = out + ((size_t)c * NF + mbase) * NXTOT + n;

    if (f0 + 15 < NF) {
        // Full tile (wave-uniform branch): no per-row guards at all.
#pragma unroll
        for (int rr = 0; rr < 8; ++rr) {
            const unsigned char mb = mp[(size_t)rr * NXTOT];
            __builtin_nontemporal_store(mb ? acc[rr] : 0.0f, &op[(size_t)rr * NXTOT]);
        }
    } else {
        // Boundary tile (f0 = 192): rows valid iff mbase < NF, i.e. half == 0;
        // then all 8 rows (192..199) are valid -> one exec toggle total.
        if (mbase < NF) {
#pragma unroll
            for (int rr = 0; rr < 8; ++rr) {
                const unsigned char mb = mp[(size_t)rr * NXTOT];
                __builtin_nontemporal_store(mb ? acc[rr] : 0.0f, &op[(size_t)rr * NXTOT]);
            }
        }
    }
}

// ---------------------------------------------------------------------------
extern "C" void kernel_launch(void* const* d_in, const int* in_sizes, int n_in,
                              void* d_out, int out_size, void* d_ws, size_t ws_size,
                              hipStream_t stream) {
    (void)in_sizes; (void)n_in; (void)out_size; (void)ws_size;
    const float*         x     = (const float*)d_in[0];          // (30,256,256,2)
    const float*         csmT  = (const float*)d_in[1];          // (8,256,256,2)
    const float*         VT    = (const float*)d_in[2];          // (30,200)
    const unsigned char* maskT = (const unsigned char*)d_in[3];  // (200,131072) bool8
    float* out = (float*)d_out;                                  // (8,200,131072)

    // Yr lives in workspace, K-padded: 8ch * 32 * 131072 floats = 134.2 MB.
    float* Yr = (float*)d_ws;
    // Pass-1 transposed intermediate (125.8 MB) reuses the (839 MB) output
    // buffer; it is fully consumed by pass 2 before pass 3 overwrites d_out.
    float* scratchT = out;

    dim3 fblock(256);
    dim3 fgrid(NXS / 2, NIMG);
    hipLaunchKernelGGL(fft_rows_kernel, fgrid, fblock, 0, stream, x, csmT, scratchT);
    hipLaunchKernelGGL(zero_pad_kernel, dim3(2048), fblock, 0, stream, Yr);
    hipLaunchKernelGGL(fft_cols_kernel, fgrid, fblock, 0, stream, scratchT, Yr);

    dim3 gblock(256);
    dim3 ggrid(NXTOT / 128, (NF + 15) / 16, NCH);
    hipLaunchKernelGGL(proj_mask_kernel, ggrid, gblock, 0, stream, Yr, VT, maskT, out);
}